// volumeRender_19602230739394
// MI455X (gfx1250) — compile-verified
//
#include <hip/hip_runtime.h>
#include <stdint.h>

#define E_FAR_F        1.0e10f
#define ATT_THRESH_F   0.001f
#define EPS_F          1e-10f
#define LOG2E_F        1.4426950408889634f

#define WAVES_PER_BLOCK 8
#define WAVE            32
#define NBUF            3                 // prefetch distance 2
#define BUF_BYTES       (WAVE * 16)       // 32 samples x float4

// ---- CDNA5 async global->LDS copy (async data-mover path) -----------------
// GLOBAL_LOAD_ASYNC_TO_LDS_B128: per-lane 16B copy, tracked by ASYNCcnt.
// rad_field is a read-once 201MB stream -> non-temporal load hint.
__device__ __forceinline__ void async_copy_b128(uint32_t lds_byte_addr,
                                                const void* gaddr) {
    asm volatile("global_load_async_to_lds_b128 %0, %1, off th:TH_LOAD_NT"
                 :
                 : "v"(lds_byte_addr), "v"(gaddr)
                 : "memory");
}

#define S_WAIT_ASYNCCNT(n) asm volatile("s_wait_asynccnt " #n ::: "memory")
#define S_WAIT_DSCNT0()    asm volatile("s_wait_dscnt 0x0"    ::: "memory")

__global__ __launch_bounds__(WAVES_PER_BLOCK * WAVE)
void volume_render_kernel(const float* __restrict__ rad_field,  // [N,S,4]
                          const float* __restrict__ depthVal,   // [N,S]
                          const float* __restrict__ ray_dir,    // [N,3]
                          float* __restrict__ out,
                          int N, int S) {
    // Triple-buffered staging: 3 x 32 samples x 16B per wave = 1.5 KB/wave.
    __shared__ float4 stage[WAVES_PER_BLOCK][NBUF][WAVE];

    const int wave = threadIdx.x >> 5;
    const int lane = threadIdx.x & (WAVE - 1);
    const int ray  = blockIdx.x * WAVES_PER_BLOCK + wave;
    if (ray >= N) return;

    // ||ray_dir||, with log2(e) folded in for the v_exp_f32 path.
    const float dx = ray_dir[(size_t)ray * 3 + 0];
    const float dy = ray_dir[(size_t)ray * 3 + 1];
    const float dz = ray_dir[(size_t)ray * 3 + 2];
    const float nrm   = sqrtf(dx * dx + dy * dy + dz * dz);
    const float nrml2 = nrm * LOG2E_F;

    // Output layout (flat, reference return order).
    const size_t Nz = (size_t)N, NS = (size_t)N * (size_t)S;
    float* color  = out;                 // [N,3]
    float* depthm = out + 3 * Nz;        // [N]
    float* wout   = out + 4 * Nz;        // [N,S]
    float* mout   = wout + NS;           // [N,S]
    float* accm   = mout + NS;           // [N]
    float* dispm  = accm + Nz;           // [N]

    const float* rf = rad_field + (size_t)ray * (size_t)S * 4;
    const float* dv = depthVal  + (size_t)ray * (size_t)S;

    // LDS byte address of this lane's slot in buffer 0 (flat addr low 32 bits
    // == LDS segment offset on gfx1250); buffers are contiguous at +512B.
    const uint32_t lds_b0 = (uint32_t)(uintptr_t)&stage[wave][0][lane];

    const int nchunk = S >> 5;  // S = 192 -> 6 chunks of 32

    // Prime the pipeline two chunks deep.
    async_copy_b128(lds_b0, rf + (size_t)lane * 4);
    if (nchunk > 1)
        async_copy_b128(lds_b0 + BUF_BYTES, rf + (size_t)(WAVE + lane) * 4);

    int rb = 0;        // buffer to read this iteration
    int ib = 2;        // buffer to fill this iteration
    float carry = 1.0f;
    float cr = 0.f, cg = 0.f, cb = 0.f, dsum = 0.f, asum = 0.f;

    for (int c = 0; c < nchunk; ++c) {
        const int s = (c << 5) + lane;

        if (c + 2 < nchunk) {
            // Drain prior ds reads before recycling their buffer (DS and
            // ASYNC LDS traffic are mutually unordered), then prefetch chunk
            // c+2 and wait until only 2 loads remain in flight => chunk c
            // has landed (async loads complete in order).
            S_WAIT_DSCNT0();
            async_copy_b128(lds_b0 + (uint32_t)ib * BUF_BYTES,
                            rf + (size_t)(s + 2 * WAVE) * 4);
            S_WAIT_ASYNCCNT(0x2);
        } else if (c + 1 < nchunk) {
            S_WAIT_ASYNCCNT(0x1);
        } else {
            S_WAIT_ASYNCCNT(0x0);
        }

        const float4 rv = stage[wave][rb][lane];      // ds_load_b128

        const float d = dv[s];
        float diff = (s == S - 1) ? E_FAR_F : (dv[s + 1] - d);

        const float sig   = fmaxf(rv.w, 0.0f);        // relu(sigma)
        // alpha = 1 - exp(-sig*diff*||dir||)  via single v_exp_f32
        const float alpha = 1.0f - __builtin_amdgcn_exp2f(-(sig * (diff * nrml2)));
        const float t     = 1.0f - alpha + EPS_F;

        // Inclusive product scan across the wave (Hillis-Steele, 5 steps).
        float p = t;
        #pragma unroll
        for (int off = 1; off < WAVE; off <<= 1) {
            const float q = __shfl_up(p, off, WAVE);
            if (lane >= off) p *= q;
        }
        // Exclusive product for this lane + inter-chunk carry.
        float excl = __shfl_up(p, 1, WAVE);
        if (lane == 0) excl = 1.0f;
        const float T = carry * excl;                 // T_i
        const float w = alpha * T;                    // weight
        const float m = (T > ATT_THRESH_F) ? 1.0f : 0.0f;
        // Whole-chunk product broadcast via readlane (no ds_bpermute).
        carry *= __uint_as_float(
            __builtin_amdgcn_readlane(__float_as_uint(p), WAVE - 1));

        // Write-once streams -> non-temporal stores.
        __builtin_nontemporal_store(w, &wout[(size_t)ray * S + s]);
        __builtin_nontemporal_store(m, &mout[(size_t)ray * S + s]);

        cr   = fmaf(w, rv.x, cr);
        cg   = fmaf(w, rv.y, cg);
        cb   = fmaf(w, rv.z, cb);
        dsum = fmaf(w, d, dsum);
        asum += w;

        rb = (rb == NBUF - 1) ? 0 : rb + 1;
        ib = (ib == NBUF - 1) ? 0 : ib + 1;
    }

    // Butterfly reduction across the wave.
    #pragma unroll
    for (int off = WAVE / 2; off >= 1; off >>= 1) {
        cr   += __shfl_xor(cr,   off, WAVE);
        cg   += __shfl_xor(cg,   off, WAVE);
        cb   += __shfl_xor(cb,   off, WAVE);
        dsum += __shfl_xor(dsum, off, WAVE);
        asum += __shfl_xor(asum, off, WAVE);
    }

    if (lane == 0) {
        const float bg = 1.0f - asum;
        color[(size_t)ray * 3 + 0] = cr + bg;
        color[(size_t)ray * 3 + 1] = cg + bg;
        color[(size_t)ray * 3 + 2] = cb + bg;
        depthm[ray] = dsum;
        accm[ray]   = asum;

        // display = 1/max(EPS, depth/acc), NaN -> 0 (jnp.maximum propagates
        // NaN; fmaxf would not, hence the explicit self-compare).
        const float ratio = dsum / asum;
        float disp;
        if (ratio != ratio) {
            disp = 0.0f;
        } else {
            disp = 1.0f / fmaxf(EPS_F, ratio);
        }
        dispm[ray] = disp;
    }
}

extern "C" void kernel_launch(void* const* d_in, const int* in_sizes, int n_in,
                              void* d_out, int out_size, void* d_ws, size_t ws_size,
                              hipStream_t stream) {
    const float* rad_field = (const float*)d_in[0];
    const float* depthVal  = (const float*)d_in[1];
    const float* ray_dir   = (const float*)d_in[2];

    const int N = in_sizes[2] / 3;          // 65536 rays
    const int S = in_sizes[1] / N;          // 192 samples (multiple of 32)

    const int blocks = (N + WAVES_PER_BLOCK - 1) / WAVES_PER_BLOCK;
    volume_render_kernel<<<blocks, WAVES_PER_BLOCK * WAVE, 0, stream>>>(
        rad_field, depthVal, ray_dir, (float*)d_out, N, S);
}